// TextureLoss_33904471834626
// MI455X (gfx1250) — compile-verified
//
#include <hip/hip_runtime.h>
#include <stdint.h>

#define H_IMG 1024
#define W_IMG 1024
#define NBATCH 8
#define TILE 24          // output tile (24x24), gray region = 32x32
#define GDIM 32
#define GS 33            // padded LDS stride for 32x32 arrays
#define MDIM 26          // mag region = TILE + 2 halo
#define MS 27
#define NT 43            // ceil(1024/24)

typedef float v2f __attribute__((ext_vector_type(2)));
typedef float v8f __attribute__((ext_vector_type(8)));

// 5-tap Gaussian, sigma=1 (matches numpy float32 build); only used to fill LDS table
__device__ __forceinline__ float band5(int d) {
    float r = 0.0f;
    r = (d == 0) ? 0.05448868f : r;
    r = (d == 1) ? 0.24420136f : r;
    r = (d == 2) ? 0.40261996f : r;
    r = (d == 3) ? 0.24420136f : r;
    r = (d == 4) ? 0.05448868f : r;
    return r;
}

// ---- LDS arena layout (bytes). chan region is overlaid by blur/mag/dir/red
// (chan dead after grayscale; barriers separate the lifetimes).
#define OFF_CHAN  0          // 2*3*1024 f32 = 24576 B   (stage A/B)
#define OFF_BLUR  0          // 2*1056  f32 =  8448 B   (stage D..E)
#define OFF_MAG   8448       // 2*702   f32 =  5616 B   (stage E..F)
#define OFF_DIR   14064      // 2*702   i32 =  5616 B   (stage E..F)
#define OFF_RED   19680      // 256     f32 =  1024 B   (reduction)
#define OFF_GRAY  24576      // 2*1056  f32 =  8448 B
#define OFF_BH    33024      // 2*1056  f32 =  8448 B
#define OFF_WH    41472      // 1056    f32 =  4224 B   (banded Gaussian matrix)
#define SMEM_BYTES 45696     // 44.6 KB -> 7 workgroups / WGP

__global__ __launch_bounds__(256) void edge_loss_kernel(
    const float* __restrict__ op, const float* __restrict__ gt,
    float* __restrict__ partial)
{
    __shared__ __align__(16) char smem_raw[SMEM_BYTES];
    float* chan = (float*)(smem_raw + OFF_CHAN);   // [(img*3+c)*1024 + e]
    float* blur = (float*)(smem_raw + OFF_BLUR);   // [img*1056 + r*GS + c]
    float* magp = (float*)(smem_raw + OFF_MAG);    // [img*702 + i*MS + j]  (squared mag)
    int*   dirp = (int*)  (smem_raw + OFF_DIR);    // [img*702 + i*MS + j]  (axis 0..3)
    float* redp = (float*)(smem_raw + OFF_RED);
    float* gray = (float*)(smem_raw + OFF_GRAY);
    float* bh   = (float*)(smem_raw + OFF_BH);
    float* wh   = (float*)(smem_raw + OFF_WH);     // wh[ci*GS+co] = g[ci-co]

    const int tid  = threadIdx.x;
    const int lane = tid & 31;
    const int wv   = tid >> 5;
    const int x0   = blockIdx.x * TILE;
    const int y0   = blockIdx.y * TILE;
    const int n    = blockIdx.z;

    const float* bases[2] = {op, gt};

    // ---- Stage A: async global->LDS staging of RGB tiles (reflect halo-4) ----
    #pragma unroll
    for (int img = 0; img < 2; ++img) {
        const float* base = bases[img] + (size_t)n * 3 * H_IMG * W_IMG;
        #pragma unroll
        for (int c = 0; c < 3; ++c) {
            const float* cb = base + (size_t)c * H_IMG * W_IMG;
            #pragma unroll
            for (int r = 0; r < 4; ++r) {
                int e  = r * 256 + tid;          // 0..1023
                int ly = e >> 5, lx = e & 31;
                int Y = y0 - 4 + ly;
                Y = (Y < 0) ? -Y : ((Y >= H_IMG) ? (2 * H_IMG - 2 - Y) : Y);
                int X = x0 - 4 + lx;
                X = (X < 0) ? -X : ((X >= W_IMG) ? (2 * W_IMG - 2 - X) : X);
                const float* gaddr = cb + (size_t)Y * W_IMG + X;
                uint32_t laddr = (uint32_t)(uintptr_t)(&chan[(img * 3 + c) * 1024 + e]);
                asm volatile("global_load_async_to_lds_b32 %0, %1, off"
                             :: "v"(laddr), "v"(gaddr) : "memory");
            }
        }
    }

    // fill banded Gaussian operand matrix while the async DMA is in flight
    for (int r = 0; r < 4; ++r) {
        int idx = r * 256 + tid;                 // 0..1023
        int ci = idx >> 5, co = idx & 31;
        wh[ci * GS + co] = band5(ci - co);
    }

    asm volatile("s_wait_asynccnt 0" ::: "memory");
    __syncthreads();

    // ---- Stage B: grayscale ----
    for (int r = 0; r < 8; ++r) {
        int img = r >> 2;
        int e   = (r & 3) * 256 + tid;
        int ly  = e >> 5, lx = e & 31;
        float g = 0.299f * chan[(img * 3 + 0) * 1024 + e]
                + 0.587f * chan[(img * 3 + 1) * 1024 + e]
                + 0.114f * chan[(img * 3 + 2) * 1024 + e];
        gray[img * (GDIM * GS) + ly * GS + lx] = g;
    }
    __syncthreads();

    // per-wave WMMA tile assignment: 8 waves = 2 images x (2x2) 16x16 tiles
    const int w_img = wv >> 2;
    const int w_t   = wv & 3;
    const int m0    = (w_t >> 1) * 16;
    const int n0    = (w_t & 1) * 16;
    const int mrow  = lane & 15;
    const int kh    = (lane >> 4) * 2;    // K offset for this lane half

    // ---- Stage C: BH = GRAY x Wh  (horizontal blur as WMMA f32 16x16x4) ----
    {
        v8f acc = {0.f,0.f,0.f,0.f,0.f,0.f,0.f,0.f};
        int slo = n0 >> 2;
        int shi = (n0 + 19) >> 2; if (shi > 7) shi = 7;   // band => skip zero K-steps
        int co  = n0 + mrow;                               // B column = N index
        const float* g0 = &gray[w_img * (GDIM * GS) + (m0 + mrow) * GS];
        for (int s = slo; s <= shi; ++s) {
            int k0 = s * 4;
            v2f a, b;
            a.x = g0[k0 + kh];
            a.y = g0[k0 + kh + 1];
            b.x = wh[(k0 + kh) * GS + co];
            b.y = wh[(k0 + kh + 1) * GS + co];
            acc = __builtin_amdgcn_wmma_f32_16x16x4_f32(
                      false, a, false, b, (short)0, acc, false, false);
        }
        for (int rr = 0; rr < 8; ++rr)
            bh[w_img * (GDIM * GS) + (m0 + rr + 8 * (lane >> 4)) * GS + n0 + mrow] = acc[rr];
    }
    __syncthreads();

    // ---- Stage D: BLUR = Wv x BH; Wv[m][k] = g[k-m] = wh[k*GS + m] ----
    {
        v8f acc = {0.f,0.f,0.f,0.f,0.f,0.f,0.f,0.f};
        int slo = m0 >> 2;
        int shi = (m0 + 19) >> 2; if (shi > 7) shi = 7;
        int ro  = m0 + mrow;                               // A row = M index
        for (int s = slo; s <= shi; ++s) {
            int k0 = s * 4;
            v2f a, b;
            a.x = wh[(k0 + kh) * GS + ro];
            a.y = wh[(k0 + kh + 1) * GS + ro];
            b.x = bh[w_img * (GDIM * GS) + (k0 + kh) * GS + n0 + mrow];
            b.y = bh[w_img * (GDIM * GS) + (k0 + kh + 1) * GS + n0 + mrow];
            acc = __builtin_amdgcn_wmma_f32_16x16x4_f32(
                      false, a, false, b, (short)0, acc, false, false);
        }
        for (int rr = 0; rr < 8; ++rr)
            blur[w_img * (GDIM * GS) + (m0 + rr + 8 * (lane >> 4)) * GS + n0 + mrow] = acc[rr];
    }
    __syncthreads();

    // ---- Stage E: Sobel (clamp pad) -> squared magnitude + quantized axis ----
    // octant via tan(22.5)=sqrt(2)-1, tan(67.5)=sqrt(2)+1; no atan2/sqrt needed.
    for (int r = 0; r < 6; ++r) {
        int item = r * 256 + tid;
        if (item < 2 * MDIM * MDIM) {
            int img = item >= MDIM * MDIM;
            int e   = item - img * MDIM * MDIM;
            int i   = e / MDIM, j = e % MDIM;
            int Y   = y0 - 1 + i, X = x0 - 1 + j;
            const float* bl = &blur[img * (GDIM * GS)];
            float b[3][3];
            #pragma unroll
            for (int dy = -1; dy <= 1; ++dy) {
                #pragma unroll
                for (int dx = -1; dx <= 1; ++dx) {
                    int Yc = Y + dy; Yc = Yc < 0 ? 0 : (Yc > H_IMG - 1 ? H_IMG - 1 : Yc);
                    int Xc = X + dx; Xc = Xc < 0 ? 0 : (Xc > W_IMG - 1 ? W_IMG - 1 : Xc);
                    b[dy + 1][dx + 1] = bl[(Yc - (y0 - 2)) * GS + (Xc - (x0 - 2))];
                }
            }
            float gx = (b[0][2] + 2.f * b[1][2] + b[2][2])
                     - (b[0][0] + 2.f * b[1][0] + b[2][0]);
            float gy = (b[2][0] + 2.f * b[2][1] + b[2][2])
                     - (b[0][0] + 2.f * b[0][1] + b[0][2]);
            float msq = gx * gx + gy * gy + 1e-6f;
            float ax = fabsf(gx), ay = fabsf(gy);
            int axis = (ay <= 0.41421356f * ax) ? 0
                     : ((ay >= 2.4142137f * ax) ? 2
                     : (((gx >= 0.f) == (gy >= 0.f)) ? 1 : 3));
            magp[img * (MDIM * MS) + i * MS + j] = msq;
            dirp[img * (MDIM * MS) + i * MS + j] = axis;
        }
    }
    __syncthreads();

    // ---- Stage F: directional NMS (zero pad) + threshold + |diff| ----
    // axis -> offset; d and d+4 are NMS-equivalent, so 4 axes suffice.
    const int DY4[4] = {0, -1, -1, -1};
    const int DX4[4] = {1,  1,  0, -1};
    float accum = 0.0f;
    for (int r = 0; r < 3; ++r) {
        int item = r * 256 + tid;
        if (item < TILE * TILE) {
            int py = item / TILE, px = item % TILE;
            int Y = y0 + py, X = x0 + px;
            if (Y < H_IMG && X < W_IMG) {
                float ed[2];
                #pragma unroll
                for (int img = 0; img < 2; ++img) {
                    const float* mg = &magp[img * (MDIM * MS)];
                    int   mi  = (py + 1) * MS + (px + 1);
                    float m0v = mg[mi];
                    int   d   = dirp[img * (MDIM * MS) + mi];
                    int dy = DY4[d], dx = DX4[d];
                    int Yp = Y + dy, Xp = X + dx;
                    float mp = (Yp >= 0 && Yp < H_IMG && Xp >= 0 && Xp < W_IMG)
                               ? mg[(py + 1 + dy) * MS + (px + 1 + dx)] : 0.0f;
                    int Yn = Y - dy, Xn = X - dx;
                    float mn = (Yn >= 0 && Yn < H_IMG && Xn >= 0 && Xn < W_IMG)
                               ? mg[(py + 1 - dy) * MS + (px + 1 - dx)] : 0.0f;
                    // squared-domain compare == magnitude compare (monotone)
                    bool ismax = (m0v > mp) && (m0v > mn);
                    ed[img] = (ismax && (m0v > 0.01f)) ? 1.0f : 0.0f;  // (0.1)^2
                }
                accum += fabsf(ed[0] - ed[1]);
            }
        }
    }

    // ---- block reduction (exact: integer-valued fp32) ----
    redp[tid] = accum;
    __syncthreads();
    for (int s2 = 128; s2 > 0; s2 >>= 1) {
        if (tid < s2) redp[tid] += redp[tid + s2];
        __syncthreads();
    }
    if (tid == 0) {
        int bl = (blockIdx.z * gridDim.y + blockIdx.y) * gridDim.x + blockIdx.x;
        partial[bl] = redp[0];
    }
}

__global__ __launch_bounds__(256) void reduce_kernel(
    const float* __restrict__ partial, int nvals, float* __restrict__ out)
{
    __shared__ float red[256];
    int tid = threadIdx.x;
    float s = 0.0f;
    for (int i = tid; i < nvals; i += 256) s += partial[i];  // fixed order: deterministic
    red[tid] = s;
    __syncthreads();
    for (int st = 128; st > 0; st >>= 1) {
        if (tid < st) red[tid] += red[tid + st];
        __syncthreads();
    }
    if (tid == 0) out[0] = red[0] * (1.0f / (8.0f * 1024.0f * 1024.0f));
}

extern "C" void kernel_launch(void* const* d_in, const int* in_sizes, int n_in,
                              void* d_out, int out_size, void* d_ws, size_t ws_size,
                              hipStream_t stream) {
    const float* op = (const float*)d_in[0];
    const float* gt = (const float*)d_in[1];
    float* partial  = (float*)d_ws;          // NT*NT*NBATCH floats = ~59 KB
    dim3 grid(NT, NT, NBATCH), block(256);
    edge_loss_kernel<<<grid, block, 0, stream>>>(op, gt, partial);
    reduce_kernel<<<1, 256, 0, stream>>>(partial, NT * NT * NBATCH, (float*)d_out);
}